// LogSlaterDeterminant_62268435857710
// MI455X (gfx1250) — compile-verified
//
#include <hip/hip_runtime.h>
#include <math.h>

typedef float v8f __attribute__((ext_vector_type(8)));
typedef float v2f __attribute__((ext_vector_type(2)));

#define S_DIM 256
#define F_DIM 64
#define LDSS  65   // padded LDS row stride (65 mod 64 == 1 -> conflict-free columns)

__global__ __launch_bounds__(32)
void logslater_wmma_kernel(const int* __restrict__ nocc,
                           const float* __restrict__ Mr,
                           const float* __restrict__ Mi,
                           float* __restrict__ out, int B)
{
    __shared__ float Ar[F_DIM * LDSS];
    __shared__ float Ai[F_DIM * LDSS];
    __shared__ int   R[F_DIM];

    const int b    = blockIdx.x;
    const int lane = threadIdx.x;          // 0..31, wave32
    const int lh   = lane >> 4;            // 0 for lanes 0-15, 1 for 16-31
    const int ln   = lane & 15;

    // ---- 1) occupied orbital indices (stable ascending, like argsort(-n)) ----
    const int* nb = nocc + (size_t)b * S_DIM;
    int nv[8];
    int cnt = 0;
    #pragma unroll
    for (int t = 0; t < 8; ++t) { nv[t] = nb[lane * 8 + t]; cnt += (nv[t] != 0); }
    int pre = cnt;
    #pragma unroll
    for (int off = 1; off < 32; off <<= 1) {
        int v = __shfl_up(pre, off, 32);
        if (lane >= off) pre += v;
    }
    pre -= cnt;                            // exclusive prefix over lanes
    {
        int pos = pre;
        #pragma unroll
        for (int t = 0; t < 8; ++t)
            if (nv[t] != 0) R[pos++] = lane * 8 + t;
    }
    __syncthreads();

    // ---- 2) gather A = M[R] into LDS (split real/imag) ----
    for (int idx = lane; idx < F_DIM * F_DIM; idx += 32) {
        int i = idx >> 6, j = idx & 63;
        int s = R[i];
        Ar[i * LDSS + j] = Mr[s * F_DIM + j];
        Ai[i * LDSS + j] = Mi[s * F_DIM + j];
    }
    __syncthreads();

    float logabs = 0.0f;
    float phase  = 0.0f;
    int   nswap  = 0;

    // ---- 3) blocked LU with partial pivoting, NB = 16 ----
    for (int kb = 0; kb < F_DIM; kb += 16) {
        // ----- panel factorization (columns kb..kb+15, rows kb..63) -----
        for (int k = kb; k < kb + 16; ++k) {
            // pivot: argmax |A[i][k]|^2 over i in [k,64)
            float bv = -1.0f; int bi = k;
            for (int i = k + lane; i < F_DIM; i += 32) {
                float xr = Ar[i * LDSS + k], xi = Ai[i * LDSS + k];
                float m  = xr * xr + xi * xi;
                if (m > bv) { bv = m; bi = i; }
            }
            #pragma unroll
            for (int off = 16; off > 0; off >>= 1) {
                float ov = __shfl_xor(bv, off, 32);
                int   oi = __shfl_xor(bi, off, 32);
                if (ov > bv || (ov == bv && oi < bi)) { bv = ov; bi = oi; }
            }
            if (bi != k) {
                for (int j = lane; j < F_DIM; j += 32) {
                    float t0 = Ar[k * LDSS + j];
                    Ar[k * LDSS + j]  = Ar[bi * LDSS + j];
                    Ar[bi * LDSS + j] = t0;
                    float t1 = Ai[k * LDSS + j];
                    Ai[k * LDSS + j]  = Ai[bi * LDSS + j];
                    Ai[bi * LDSS + j] = t1;
                }
                nswap++;
            }
            __syncthreads();

            float ur = Ar[k * LDSS + k], ui = Ai[k * LDSS + k];
            float m2 = ur * ur + ui * ui;
            logabs += 0.5f * logf(m2);
            phase  += atan2f(ui, ur);
            float inv = 1.0f / m2;

            // multipliers + rank-1 update restricted to the panel columns
            for (int i = k + 1 + lane; i < F_DIM; i += 32) {
                float a0 = Ar[i * LDSS + k], a1 = Ai[i * LDSS + k];
                float lr = (a0 * ur + a1 * ui) * inv;   // a / u (complex)
                float li = (a1 * ur - a0 * ui) * inv;
                Ar[i * LDSS + k] = lr;
                Ai[i * LDSS + k] = li;
                for (int j = k + 1; j < kb + 16; ++j) {
                    float pr = Ar[k * LDSS + j], pim = Ai[k * LDSS + j];
                    Ar[i * LDSS + j] -= lr * pr - li * pim;
                    Ai[i * LDSS + j] -= lr * pim + li * pr;
                }
            }
            __syncthreads();
        }

        if (kb + 16 >= F_DIM) break;
        const int jt0 = kb + 16;
        const int W   = F_DIM - jt0;

        // ----- U12 = L11^{-1} * A12 (unit-lower triangular solve) -----
        for (int jj = lane; jj < W; jj += 32) {
            int j = jt0 + jj;
            float cr_[16], ci_[16];
            #pragma unroll
            for (int t = 0; t < 16; ++t) {
                cr_[t] = Ar[(kb + t) * LDSS + j];
                ci_[t] = Ai[(kb + t) * LDSS + j];
            }
            #pragma unroll
            for (int r = 1; r < 16; ++r) {
                float ar = cr_[r], aim = ci_[r];
                #pragma unroll
                for (int t = 0; t < r; ++t) {
                    float lr = Ar[(kb + r) * LDSS + kb + t];
                    float li = Ai[(kb + r) * LDSS + kb + t];
                    ar  -= lr * cr_[t] - li * ci_[t];
                    aim -= lr * ci_[t] + li * cr_[t];
                }
                cr_[r] = ar; ci_[r] = aim;
            }
            #pragma unroll
            for (int t = 1; t < 16; ++t) {
                Ar[(kb + t) * LDSS + j] = cr_[t];
                Ai[(kb + t) * LDSS + j] = ci_[t];
            }
        }
        __syncthreads();

        // ----- trailing update A22 -= L21 * U12 via V_WMMA_F32_16X16X4_F32 -----
        // complex: Cr += -Lr*Ur + Li*Ui ; Ci += -Lr*Ui - Li*Ur
        for (int i0 = jt0; i0 < F_DIM; i0 += 16) {
            for (int j0 = jt0; j0 < F_DIM; j0 += 16) {
                v8f cr, ci;
                #pragma unroll
                for (int v = 0; v < 8; ++v) {      // C/D layout: VGPR v -> rows v, v+8
                    cr[v] = Ar[(i0 + v + 8 * lh) * LDSS + j0 + ln];
                    ci[v] = Ai[(i0 + v + 8 * lh) * LDSS + j0 + ln];
                }
                #pragma unroll
                for (int tc = 0; tc < 4; ++tc) {   // K = 16 in chunks of 4
                    int t0 = kb + 4 * tc + 2 * lh; // lanes 16-31 hold K+2/K+3
                    v2f aLr, aLi, nLr, nLi, bUr, bUi;
                    aLr[0] = Ar[(i0 + ln) * LDSS + t0];
                    aLr[1] = Ar[(i0 + ln) * LDSS + t0 + 1];
                    aLi[0] = Ai[(i0 + ln) * LDSS + t0];
                    aLi[1] = Ai[(i0 + ln) * LDSS + t0 + 1];
                    nLr = -aLr; nLi = -aLi;        // F32 WMMA has no A/B neg bits
                    bUr[0] = Ar[t0 * LDSS + j0 + ln];
                    bUr[1] = Ar[(t0 + 1) * LDSS + j0 + ln];
                    bUi[0] = Ai[t0 * LDSS + j0 + ln];
                    bUi[1] = Ai[(t0 + 1) * LDSS + j0 + ln];
                    cr = __builtin_amdgcn_wmma_f32_16x16x4_f32(
                             false, nLr, false, bUr, (short)0, cr, false, false);
                    cr = __builtin_amdgcn_wmma_f32_16x16x4_f32(
                             false, aLi, false, bUi, (short)0, cr, false, false);
                    ci = __builtin_amdgcn_wmma_f32_16x16x4_f32(
                             false, nLr, false, bUi, (short)0, ci, false, false);
                    ci = __builtin_amdgcn_wmma_f32_16x16x4_f32(
                             false, nLi, false, bUr, (short)0, ci, false, false);
                }
                #pragma unroll
                for (int v = 0; v < 8; ++v) {
                    Ar[(i0 + v + 8 * lh) * LDSS + j0 + ln] = cr[v];
                    Ai[(i0 + v + 8 * lh) * LDSS + j0 + ln] = ci[v];
                }
            }
        }
        __syncthreads();
    }

    // ---- 4) output: log|det| + i*arg(det), phase wrapped to (-pi, pi] ----
    if (lane == 0) {
        const float PI    = 3.14159265358979323846f;
        const float TWOPI = 6.28318530717958647692f;
        float ph = phase + PI * (float)nswap;     // each row swap: det *= -1
        ph = ph - TWOPI * rintf(ph * (1.0f / TWOPI));
        out[2 * b + 0] = logabs;
        out[2 * b + 1] = ph;
    }
}

extern "C" void kernel_launch(void* const* d_in, const int* in_sizes, int n_in,
                              void* d_out, int out_size, void* d_ws, size_t ws_size,
                              hipStream_t stream) {
    const int*   n  = (const int*)d_in[0];    // (B, 256) int32 occupation
    const float* Mr = (const float*)d_in[1];  // (256, 64) f32
    const float* Mi = (const float*)d_in[2];  // (256, 64) f32
    float* out = (float*)d_out;               // B complex64 = 2B floats
    int B = in_sizes[0] / S_DIM;
    hipLaunchKernelGGL(logslater_wmma_kernel, dim3(B), dim3(32), 0, stream,
                       n, Mr, Mi, out, B);
}